// YOLOLoss_13383118094573
// MI455X (gfx1250) — compile-verified
//
#include <hip/hip_runtime.h>
#include <stdint.h>

#define NCLS 80
#define CPC  85      // channels per anchor (5 + 80)
#define MAXT 128
#define BATCH 16

#define GT_THREADS 1024   // 32 waves per batch image
#define GT_CHUNK   4      // cells per thread per memory round-trip

#if defined(__has_builtin)
# if __has_builtin(__builtin_amdgcn_global_load_async_to_lds_b128)
#  define USE_ASYNC_LDS 1
# endif
#endif

typedef int v4i_async __attribute__((vector_size(4 * sizeof(int))));
typedef __attribute__((address_space(1))) v4i_async* g_v4i_ptr;
typedef __attribute__((address_space(3))) v4i_async* l_v4i_ptr;

__device__ __forceinline__ float softplusf(float x) {
    // stable: max(x,0) + log1p(exp(-|x|))
    return fmaxf(x, 0.0f) + log1pf(__expf(-fabsf(x)));
}
__device__ __forceinline__ float bcef(float x, float t) {
    return softplusf(x) - x * t;
}
__device__ __forceinline__ float sigmoidf_(float x) {
    return 1.0f / (1.0f + __expf(-x));
}

// ---------------------------------------------------------------------------
// Zero accumulators + counts (d_ws is poisoned 0xAA by the harness).
// ---------------------------------------------------------------------------
__global__ void yolo_zero_ws(float* __restrict__ acc, int* __restrict__ cnts) {
    int t = threadIdx.x;
    if (t < 4)  acc[t]  = 0.0f;
    if (t < 3 * BATCH) cnts[t] = 0;
}

// ---------------------------------------------------------------------------
// Ordered compaction of true boxes (obj==1 cells, raster order, cap 128).
// One 32-wave block per batch image. Each round: 4096 flags loaded in one
// latency exposure (4 independent loads/thread), then 4 ballot+block-scan
// sub-chunks entirely in-core. Matches the reference's stable argsort of the
// 0/1 mask exactly, and is fully deterministic.
// ---------------------------------------------------------------------------
__global__ void yolo_gather_true(const float* __restrict__ yt,
                                 float* __restrict__ tb,
                                 int* __restrict__ cnt, int S) {
    const int b    = blockIdx.x;
    const int tid  = threadIdx.x;
    const int lane = tid & 31;
    const int wv   = tid >> 5;              // wave id 0..31
    const int N    = S * S * 3;
    const float* ytb = yt + (size_t)b * N * CPC;
    float* tbb = tb + (size_t)b * MAXT * 4;

    __shared__ int wcnt[32];
    __shared__ int wpre[32];
    __shared__ int stot;

    int base = 0;
    const int round = GT_THREADS * GT_CHUNK;    // 4096 cells / round-trip
    for (int p0 = 0; p0 < N; p0 += round) {
        // one memory round-trip: 4 independent flag loads per thread
        float m[GT_CHUNK];
        int   p[GT_CHUNK];
#pragma unroll
        for (int k = 0; k < GT_CHUNK; ++k) {
            p[k] = p0 + k * GT_THREADS + tid;
            m[k] = (p[k] < N) ? ytb[(size_t)p[k] * CPC + 4] : 0.0f;
        }
#pragma unroll
        for (int k = 0; k < GT_CHUNK; ++k) {
            const bool flag = (m[k] == 1.0f);
            const unsigned bal = (unsigned)__ballot(flag);   // wave32 mask
            if (lane == 0) wcnt[wv] = __popc(bal);
            __syncthreads();                 // wcnt ready; prior wpre/stot uses done
            if (wv == 0) {                   // wave-0 scans the 32 wave counts
                int c = wcnt[lane];
#pragma unroll
                for (int o = 1; o < 32; o <<= 1) {
                    int t = __shfl_up(c, o);
                    if (lane >= o) c += t;
                }
                wpre[lane] = c - wcnt[lane]; // exclusive prefix
                if (lane == 31) stot = c;    // chunk total
            }
            __syncthreads();
            const int idx = base + wpre[wv] + __popc(bal & ((1u << lane) - 1u));
            if (flag && idx < MAXT) {
                const float* c4 = ytb + (size_t)p[k] * CPC;
                tbb[idx * 4 + 0] = c4[0]; tbb[idx * 4 + 1] = c4[1];
                tbb[idx * 4 + 2] = c4[2]; tbb[idx * 4 + 3] = c4[3];
            }
            base += stot;
        }
    }
    if (tid == 0) cnt[b] = (base < MAXT) ? base : MAXT;
}

// ---------------------------------------------------------------------------
// Per-scale loss. grid = (ceil(3*S*S/256), B), block = 256 (8 waves).
// Consecutive threads -> consecutive hw within one (b, anchor, channel) plane
// => fully coalesced fm loads. Class channels (5..84) gated on wave-any(obj).
// True boxes staged into LDS via async global->LDS b128 (ASYNCcnt path).
// acc layout: [xy, wh, conf, cls] raw sums (all scales combined).
// ---------------------------------------------------------------------------
__global__ void yolo_scale_loss(const float* __restrict__ fm,
                                const float* __restrict__ yt,
                                const float* __restrict__ tb,
                                const int*   __restrict__ cnt,
                                float* __restrict__ acc, int S,
                                float aw0, float ah0, float aw1, float ah1,
                                float aw2, float ah2) {
    __shared__ float tbL[MAXT * 4];
    __shared__ float blk[4];

    const int tid   = threadIdx.x;
    const int b     = blockIdx.y;
    const int HW    = S * S;
    const int cells = 3 * HW;
    const int n     = cnt[b];
    const float* tbg = tb + (size_t)b * MAXT * 4;

#ifdef USE_ASYNC_LDS
    // CDNA5 async global->LDS copy (ASYNCcnt path): 128 boxes x 16B.
    if (tid < MAXT) {
        __builtin_amdgcn_global_load_async_to_lds_b128(
            (g_v4i_ptr)(tbg + (size_t)tid * 4),
            (l_v4i_ptr)(&tbL[tid * 4]),
            0, 0);
    }
# if defined(__has_builtin) && __has_builtin(__builtin_amdgcn_s_wait_asynccnt)
    __builtin_amdgcn_s_wait_asynccnt(0);
# else
    asm volatile("s_wait_asynccnt 0" ::: "memory");
# endif
#else
    if (tid < MAXT) {
        float4 v = ((const float4*)tbg)[tid];
        *(float4*)&tbL[tid * 4] = v;
    }
#endif
    if (tid < 4) blk[tid] = 0.0f;
    __syncthreads();

    float v0 = 0.f, v1 = 0.f, v2 = 0.f, v3 = 0.f;   // xy, wh, conf, cls

    const int g = blockIdx.x * blockDim.x + tid;
    const bool live = (g < cells);
    if (live) {
        const int a  = g / HW;
        const int hw = g - a * HW;
        const int h  = hw / S;
        const int w  = hw - h * S;
        const float fd = (float)S;                  // H == W on every scale

        const size_t fbase = ((size_t)b * 255 + (size_t)a * CPC) * HW + hw;
        const float r0 = fm[fbase];
        const float r1 = fm[fbase + (size_t)HW];
        const float r2 = fm[fbase + 2 * (size_t)HW];
        const float r3 = fm[fbase + 3 * (size_t)HW];
        const float r4 = fm[fbase + 4 * (size_t)HW];

        const float* yc = yt + (((size_t)b * HW + hw) * 3 + a) * CPC;
        const float tx  = yc[0], ty = yc[1], tw = yc[2], th = yc[3];
        const float obj = yc[4];

        const float aw = (a == 0) ? aw0 : ((a == 1) ? aw1 : aw2);
        const float ah = (a == 0) ? ah0 : ((a == 1) ? ah1 : ah2);

        // xy loss: BCE(raw_xy, y*grid_size - grid)
        const float bs = 2.0f - tw * th;
        v0 = obj * bs * (bcef(r0, tx * fd - (float)w) +
                         bcef(r1, ty * fd - (float)h));

        // wh loss: (raw_wh - log(y_wh/anchor*608))^2  (only where obj==1)
        if (obj != 0.0f) {
            const float lw = __logf(tw / aw * 608.0f);
            const float lh = __logf(th / ah * 608.0f);
            const float dw = r2 - lw, dh = r3 - lh;
            v1 = obj * bs * (dw * dw + dh * dh);
        }

        // decoded prediction box
        const float px = (sigmoidf_(r0) + (float)w) / fd;
        const float py = (sigmoidf_(r1) + (float)h) / fd;
        const float pw = aw * __expf(r2) * (1.0f / 608.0f);
        const float ph = ah * __expf(r3) * (1.0f / 608.0f);
        const float pminx = px - 0.5f * pw, pmaxx = px + 0.5f * pw;
        const float pminy = py - 0.5f * ph, pmaxy = py + 0.5f * ph;
        const float pa = pw * ph;

        const bool anyobj = __any(obj != 0.0f);
        if (anyobj && obj != 0.0f) {
            // pull class logits toward L2/WGP$ while the IoU loop runs
            __builtin_prefetch(&fm[fbase + 5 * (size_t)HW], 0, 0);
        }

        // best IoU vs true boxes (LDS broadcast reads)
        float best = 0.0f;
        for (int j = 0; j < n; ++j) {
            const float bx = tbL[j * 4 + 0], by = tbL[j * 4 + 1];
            const float bw = tbL[j * 4 + 2], bh = tbL[j * 4 + 3];
            const float iw = fmaxf(fminf(pmaxx, bx + 0.5f * bw) -
                                   fmaxf(pminx, bx - 0.5f * bw), 0.0f);
            const float ih = fmaxf(fminf(pmaxy, by + 0.5f * bh) -
                                   fmaxf(pminy, by - 0.5f * bh), 0.0f);
            const float inter = iw * ih;
            const float iou = inter / (pa + bw * bh - inter);
            best = fmaxf(best, iou);
        }
        const float ign = (best < 0.5f) ? 1.0f : 0.0f;
        const float bc  = bcef(r4, obj);
        v2 = obj * bc + (1.0f - obj) * bc * ign;

        // class loss: only where obj==1 (0.5% of cells; ~85% of waves skip
        // the 80-channel read entirely -> big DRAM traffic cut)
        if (anyobj && obj != 0.0f) {
            float s = 0.0f;
            for (int c = 0; c < NCLS; ++c) {
                s += bcef(fm[fbase + (size_t)(5 + c) * HW], yc[5 + c]);
            }
            v3 = obj * s;
        }
    }

    // wave32 butterfly reduce -> LDS -> global atomics
    for (int o = 16; o > 0; o >>= 1) {
        v0 += __shfl_xor(v0, o);
        v1 += __shfl_xor(v1, o);
        v2 += __shfl_xor(v2, o);
        v3 += __shfl_xor(v3, o);
    }
    if ((tid & 31) == 0) {
        atomicAdd(&blk[0], v0);
        atomicAdd(&blk[1], v1);
        atomicAdd(&blk[2], v2);
        atomicAdd(&blk[3], v3);
    }
    __syncthreads();
    if (tid < 4) atomicAdd(&acc[tid], blk[tid]);
}

// ---------------------------------------------------------------------------
// Final 5-element output: [total, xy, wh, conf, cls], each sum / B.
// ---------------------------------------------------------------------------
__global__ void yolo_finalize(const float* __restrict__ acc,
                              float* __restrict__ out) {
    __builtin_amdgcn_s_wait_tensorcnt(0);   // gfx1250-specific, harmless no-op
    if (threadIdx.x == 0) {
        const float inv = 1.0f / (float)BATCH;
        const float xy = acc[0] * inv, wh = acc[1] * inv;
        const float cf = acc[2] * inv, cl = acc[3] * inv;
        out[0] = xy + wh + cf + cl;
        out[1] = xy; out[2] = wh; out[3] = cf; out[4] = cl;
    }
}

// ---------------------------------------------------------------------------
extern "C" void kernel_launch(void* const* d_in, const int* in_sizes, int n_in,
                              void* d_out, int out_size, void* d_ws, size_t ws_size,
                              hipStream_t stream) {
    const float* fm[3] = {(const float*)d_in[0], (const float*)d_in[1],
                          (const float*)d_in[2]};
    const float* yt[3] = {(const float*)d_in[3], (const float*)d_in[4],
                          (const float*)d_in[5]};
    const int S[3] = {19, 38, 76};

    // ws layout: acc[4] floats | pad | tb[3][16][128][4] floats | cnts[48] ints
    float* acc = (float*)d_ws;
    float* tbw = (float*)d_ws + 16;                      // 64B-aligned
    int*   cnts = (int*)((float*)d_ws + 16 + 3 * BATCH * MAXT * 4);

    yolo_zero_ws<<<1, 64, 0, stream>>>(acc, cnts);

    // ANCHOR_MASK: scale0 -> anchors[6..8], scale1 -> [3..5], scale2 -> [0..2]
    const float AW[3][3] = {{116.f, 156.f, 373.f}, {30.f, 62.f, 59.f}, {10.f, 16.f, 33.f}};
    const float AH[3][3] = {{ 90.f, 198.f, 326.f}, {61.f, 45.f, 119.f}, {13.f, 30.f, 23.f}};

    for (int s = 0; s < 3; ++s) {
        float* tbs = tbw + (size_t)s * BATCH * MAXT * 4;
        int*   cs  = cnts + s * BATCH;
        // gather right before the matching loss pass -> y_true stays hot in
        // the 192MB L2 for the second (main) read.
        yolo_gather_true<<<BATCH, GT_THREADS, 0, stream>>>(yt[s], tbs, cs, S[s]);

        const int cells = 3 * S[s] * S[s];
        dim3 grid((cells + 255) / 256, BATCH);
        yolo_scale_loss<<<grid, 256, 0, stream>>>(
            fm[s], yt[s], tbs, cs, acc, S[s],
            AW[s][0], AH[s][0], AW[s][1], AH[s][1], AW[s][2], AH[s][2]);
    }

    yolo_finalize<<<1, 32, 0, stream>>>(acc, (float*)d_out);
}